// StructuralAttention_84834194031238
// MI455X (gfx1250) — compile-verified
//
#include <hip/hip_runtime.h>
#include <hip/hip_bf16.h>

#define DIM  128
#define TILE 16

typedef float v2f __attribute__((ext_vector_type(2)));
typedef float v8f __attribute__((ext_vector_type(8)));

// ---------------------------------------------------------------------------
// Kernel 1: qk = x*Wqk + bqk ; v = x*Wv + bv  -- fp32 WMMA (16x16x4)
// grid = (N/16, 16): blockIdx.y 0..7 -> qk col tiles, 8..15 -> v col tiles
// one wave (32 lanes) per 16x16 output tile
// ---------------------------------------------------------------------------
__global__ void __launch_bounds__(32)
gemm_qkv_kernel(const float* __restrict__ x,
                const float* __restrict__ Wqk, const float* __restrict__ bqk,
                const float* __restrict__ Wv,  const float* __restrict__ bv,
                float* __restrict__ qk, float* __restrict__ v, int N)
{
    const int row0 = blockIdx.x * TILE;
    const int t    = blockIdx.y;
    const int col0 = (t & 7) * TILE;
    const float* W   = (t < 8) ? Wqk : Wv;
    const float* bia = (t < 8) ? bqk : bv;
    float*       out = (t < 8) ? qk  : v;

    const int lane = threadIdx.x;
    const int half = lane >> 4;      // 0: K={k0,k0+1}, 1: K={k0+2,k0+3}
    const int l16  = lane & 15;

    int arow = row0 + l16; if (arow >= N) arow = N - 1;   // clamp (N%16==0 normally)
    const float* xrow = x + (size_t)arow * DIM;

    v8f c = {0.f, 0.f, 0.f, 0.f, 0.f, 0.f, 0.f, 0.f};
    for (int k0 = 0; k0 < DIM; k0 += 4) {
        const int ka = k0 + half * 2;
        v2f a, b;
        a.x = xrow[ka];
        a.y = xrow[ka + 1];
        b.x = W[(size_t)ka       * DIM + col0 + l16];
        b.y = W[(size_t)(ka + 1) * DIM + col0 + l16];
        c = __builtin_amdgcn_wmma_f32_16x16x4_f32(false, a, false, b,
                                                  (short)0, c, false, false);
    }

    const float bb = bia[col0 + l16];
#pragma unroll
    for (int i = 0; i < 8; ++i) {
        const int r = row0 + i + half * 8;
        if (r < N) out[(size_t)r * DIM + col0 + l16] = c[i] + bb;
    }
}

// ---------------------------------------------------------------------------
// Kernel 2: per-edge score = clamp(dot(qk[dst],qk[src])/sqrt(D), -5, 5)
//           escore = exp(score); sums[dst] += escore   (one wave per edge)
// ---------------------------------------------------------------------------
__global__ void __launch_bounds__(256)
edge_score_kernel(const float* __restrict__ qk,
                  const int* __restrict__ esrc, const int* __restrict__ edst,
                  float* __restrict__ escore, float* __restrict__ sums, int E)
{
    const int lane = threadIdx.x & 31;
    const int e    = blockIdx.x * (blockDim.x >> 5) + (threadIdx.x >> 5);
    if (e >= E) return;

    const int s = esrc[e], d = edst[e];
    const float4 a = ((const float4*)(qk + (size_t)s * DIM))[lane];
    const float4 b = ((const float4*)(qk + (size_t)d * DIM))[lane];
    float p = a.x * b.x + a.y * b.y + a.z * b.z + a.w * b.w;
#pragma unroll
    for (int off = 16; off > 0; off >>= 1) p += __shfl_xor(p, off, 32);

    if (lane == 0) {
        float sc = p * 0.08838834764831845f;          // 1/sqrt(128)
        sc = fminf(5.0f, fmaxf(-5.0f, sc));
        const float ex = __expf(sc);
        escore[e] = ex;
        unsafeAtomicAdd(&sums[d], ex);
    }
}

// ---------------------------------------------------------------------------
// Kernel 3: agg[dst] += (escore/sums[dst]) * v[src]   (one wave per edge)
// ---------------------------------------------------------------------------
__global__ void __launch_bounds__(256)
edge_aggregate_kernel(const float* __restrict__ v,
                      const int* __restrict__ esrc, const int* __restrict__ edst,
                      const float* __restrict__ escore, const float* __restrict__ sums,
                      float* __restrict__ agg, int E)
{
    const int lane = threadIdx.x & 31;
    const int e    = blockIdx.x * (blockDim.x >> 5) + (threadIdx.x >> 5);
    if (e >= E) return;

    const int s = esrc[e], d = edst[e];
    const float w = escore[e] / sums[d];
    const float4 vv = ((const float4*)(v + (size_t)s * DIM))[lane];
    float* dstp = agg + (size_t)d * DIM + lane * 4;
    unsafeAtomicAdd(dstp + 0, w * vv.x);
    unsafeAtomicAdd(dstp + 1, w * vv.y);
    unsafeAtomicAdd(dstp + 2, w * vv.z);
    unsafeAtomicAdd(dstp + 3, w * vv.w);
}

// ---------------------------------------------------------------------------
// Kernel 4: y = LayerNorm(agg@Wo + bo + x) * gamma + beta
// 256 threads = 8 waves; wave w computes col tile w of the 16x128 block with
// fp32 WMMA into LDS (fused bias + residual), then 2 rows of LayerNorm each.
// ---------------------------------------------------------------------------
__global__ void __launch_bounds__(256)
out_proj_ln_kernel(const float* __restrict__ agg, const float* __restrict__ x,
                   const float* __restrict__ Wo,  const float* __restrict__ bo,
                   const float* __restrict__ gamma, const float* __restrict__ beta,
                   float* __restrict__ y, int N)
{
    __shared__ float h[TILE][DIM];

    const int row0 = blockIdx.x * TILE;
    const int wid  = threadIdx.x >> 5;   // 0..7 -> column tile
    const int lane = threadIdx.x & 31;
    const int half = lane >> 4;
    const int l16  = lane & 15;
    const int col0 = wid * TILE;

    int arow = row0 + l16; if (arow >= N) arow = N - 1;
    const float* ar = agg + (size_t)arow * DIM;

    v8f c = {0.f, 0.f, 0.f, 0.f, 0.f, 0.f, 0.f, 0.f};
    for (int k0 = 0; k0 < DIM; k0 += 4) {
        const int ka = k0 + half * 2;
        v2f a, b;
        a.x = ar[ka];
        a.y = ar[ka + 1];
        b.x = Wo[(size_t)ka       * DIM + col0 + l16];
        b.y = Wo[(size_t)(ka + 1) * DIM + col0 + l16];
        c = __builtin_amdgcn_wmma_f32_16x16x4_f32(false, a, false, b,
                                                  (short)0, c, false, false);
    }

    const float bb = bo[col0 + l16];
#pragma unroll
    for (int i = 0; i < 8; ++i) {
        const int r  = i + half * 8;
        int gr = row0 + r; if (gr >= N) gr = N - 1;
        h[r][col0 + l16] = c[i] + bb + x[(size_t)gr * DIM + col0 + l16];
    }
    __syncthreads();

    // LayerNorm: wave `wid` handles rows 2*wid and 2*wid+1
#pragma unroll
    for (int rr = 0; rr < 2; ++rr) {
        const int r = wid * 2 + rr;
        const float t0 = h[r][lane],      t1 = h[r][lane + 32];
        const float t2 = h[r][lane + 64], t3 = h[r][lane + 96];

        float sm = t0 + t1 + t2 + t3;
#pragma unroll
        for (int off = 16; off > 0; off >>= 1) sm += __shfl_xor(sm, off, 32);
        const float mu = sm * (1.0f / DIM);

        const float d0 = t0 - mu, d1 = t1 - mu, d2 = t2 - mu, d3 = t3 - mu;
        float vs = d0 * d0 + d1 * d1 + d2 * d2 + d3 * d3;
#pragma unroll
        for (int off = 16; off > 0; off >>= 1) vs += __shfl_xor(vs, off, 32);
        const float inv = rsqrtf(vs * (1.0f / DIM) + 1e-5f);

        const int gr = row0 + r;
        if (gr < N) {
            float* yr = y + (size_t)gr * DIM;
            yr[lane]      = gamma[lane]      * (d0 * inv) + beta[lane];
            yr[lane + 32] = gamma[lane + 32] * (d1 * inv) + beta[lane + 32];
            yr[lane + 64] = gamma[lane + 64] * (d2 * inv) + beta[lane + 64];
            yr[lane + 96] = gamma[lane + 96] * (d3 * inv) + beta[lane + 96];
        }
    }
}

// ---------------------------------------------------------------------------
extern "C" void kernel_launch(void* const* d_in, const int* in_sizes, int n_in,
                              void* d_out, int out_size, void* d_ws, size_t ws_size,
                              hipStream_t stream)
{
    const float* x     = (const float*)d_in[0];
    const int*   edge  = (const int*)d_in[1];     // [2, E] int32 (x64 disabled)
    const float* Wqk   = (const float*)d_in[2];
    const float* bqk   = (const float*)d_in[3];
    const float* Wv    = (const float*)d_in[4];
    const float* bv    = (const float*)d_in[5];
    const float* Wo    = (const float*)d_in[6];
    const float* bo    = (const float*)d_in[7];
    const float* gamma = (const float*)d_in[8];
    const float* beta  = (const float*)d_in[9];
    float*       y     = (float*)d_out;

    const int N = in_sizes[0] / DIM;
    const int E = in_sizes[1] / 2;
    const int* esrc = edge;
    const int* edst = edge + E;

    // scratch layout: qk | v | agg | escore | sums  (~161 MB for N=100k,E=1.6M)
    float* qk     = (float*)d_ws;
    float* v      = qk  + (size_t)N * DIM;
    float* agg    = v   + (size_t)N * DIM;
    float* escore = agg + (size_t)N * DIM;
    float* sums   = escore + E;

    hipMemsetAsync(agg,  0, (size_t)N * DIM * sizeof(float), stream);
    hipMemsetAsync(sums, 0, (size_t)N * sizeof(float), stream);

    // 1) qk / v projections (fp32 WMMA)
    dim3 g1((N + TILE - 1) / TILE, 16);
    gemm_qkv_kernel<<<g1, 32, 0, stream>>>(x, Wqk, bqk, Wv, bv, qk, v, N);

    // 2) edge scores + softmax denominators
    const int wavesPerBlock = 8;
    const int nblk = (E + wavesPerBlock - 1) / wavesPerBlock;
    edge_score_kernel<<<nblk, 256, 0, stream>>>(qk, esrc, edst, escore, sums, E);

    // 3) weighted scatter-add aggregation
    edge_aggregate_kernel<<<nblk, 256, 0, stream>>>(v, esrc, edst, escore, sums, agg, E);

    // 4) output projection + residual + layernorm (fp32 WMMA)
    out_proj_ln_kernel<<<(N + TILE - 1) / TILE, 256, 0, stream>>>(
        agg, x, Wo, bo, gamma, beta, y, N);
}